// LaplacianPositionalEncoding_31920196944059
// MI455X (gfx1250) — compile-verified
//
#include <hip/hip_runtime.h>
#include <hip/hip_bf16.h>

// Laplacian positional encoding for 4096 graphs of 64 nodes.
// One 128-thread workgroup (4 wave32) per graph. All work in LDS.
// Eigensolve = subspace iteration with CholeskyQR re-orthonormalization
// (all GEMMs on V_WMMA_F32_16X16X4_F32) + Rayleigh-Ritz (32x32 parallel
// Jacobi) + WMMA back-projection.

#define GG    4096
#define NPERN 64
#define KOUT  16
#define EPERE 128
#define MSUB  32          // subspace dimension (16 wanted + 16 guard)
#define LB    65          // LDS row stride for 64x64 B (odd -> bank friendly)
#define LX    33          // LDS row stride for 64x32 X/Y
#define LS    33          // LDS row stride for 32x32 S/W
#define PITER 12          // subspace iterations
#define JSWEEPS 6         // Jacobi sweeps on 32x32 Ritz matrix

typedef __attribute__((ext_vector_type(2))) float v2f;
typedef __attribute__((ext_vector_type(8))) float v8f;

// One 16x16 f32 C tile = A(16xK) @ B(Kx16) using V_WMMA_F32_16X16X4_F32.
// A-operand layout (ISA 7.12.2, 32-bit A 16x4): lane m=lane&15 holds row m;
//   VGPR0 = K = k+2*(lane>>4), VGPR1 = K+1.
// B-operand (4x16, rows striped across lanes): VGPR0 = row k+2*(lane>>4),
//   col = lane&15; VGPR1 = next row.
// AT: operand A is stored transposed (A[m][k] read as Asrc[k][m]).
// BT: operand B is stored transposed (B[k][n] read as Bsrc[n][k]).
template <bool AT, bool BT>
__device__ __forceinline__ v8f wmma_mm16(const float* __restrict__ A, int lda,
                                         const float* __restrict__ B, int ldb,
                                         int K, int lane) {
  v8f acc = {};
  const int mr = lane & 15;
  const int hi = (lane >> 4) & 1;
  for (int k = 0; k < K; k += 4) {
    const int k0 = k + 2 * hi;
    v2f a, b;
    if (AT) {
      a.x = A[k0 * lda + mr];
      a.y = A[(k0 + 1) * lda + mr];
    } else {
      a.x = A[mr * lda + k0];
      a.y = A[mr * lda + k0 + 1];
    }
    if (BT) {
      b.x = B[mr * ldb + k0];
      b.y = B[mr * ldb + k0 + 1];
    } else {
      b.x = B[k0 * ldb + mr];
      b.y = B[(k0 + 1) * ldb + mr];
    }
    acc = __builtin_amdgcn_wmma_f32_16x16x4_f32(
        false, a, false, b, (short)0, acc, false, false);
  }
  return acc;
}

// C/D layout: VGPR r holds M = r + 8*(lane>>4), N = lane&15.
__device__ __forceinline__ void store_tile(float* __restrict__ C, int ldc,
                                           int lane, v8f acc) {
  const int mr = lane & 15;
  const int hi = (lane >> 4) & 1;
#pragma unroll
  for (int r = 0; r < 8; ++r) C[(r + 8 * hi) * ldc + mr] = acc[r];
}

__global__ __launch_bounds__(128) void lappe_eigh_kernel(
    const int* __restrict__ ei, float* __restrict__ out) {
  __shared__ float Bm[NPERN * LB];     // adjacency -> B = 2I - L
  __shared__ float Xl[NPERN * LX];     // current orthonormal basis 64x32
  __shared__ float Yl[NPERN * LX];     // B @ X
  __shared__ float Sl[MSUB * LS];      // Gram / Ritz matrix 32x32
  __shared__ float Wl[MSUB * LS];      // L^-1 scratch, then Jacobi vectors
  __shared__ float Wsel[MSUB * KOUT];  // selected 16 columns of W
  __shared__ float dinv[NPERN];
  __shared__ float invd[MSUB];         // 1/L[k][k] from Cholesky
  __shared__ float cb[16], sb[16];
  __shared__ int pb[16], qb[16];
  __shared__ int ord[KOUT];

  const int g = blockIdx.x;
  const int tid = threadIdx.x;
  const int lane = tid & 31;
  const int wv = tid >> 5;

  // ---- zero adjacency -------------------------------------------------
  for (int i = tid; i < NPERN * LB; i += 128) Bm[i] = 0.0f;
  __syncthreads();

  // ---- accumulate edges (to_dense_adj sums duplicates -> LDS f32 atomics)
  // edge_index is [2, 2*G*EPER]; row = concat(s,d), col = concat(d,s).
  {
    const int* rowp = ei;
    const int* colp = ei + 2 * GG * EPERE;  // second row of edge_index
    int e1 = g * EPERE + tid;               // s-half, tid < 128 == EPER
    int r = rowp[e1] & (NPERN - 1);
    int c = colp[e1] & (NPERN - 1);
    atomicAdd(&Bm[r * LB + c], 1.0f);
    int e2 = GG * EPERE + g * EPERE + tid;  // d-half
    r = rowp[e2] & (NPERN - 1);
    c = colp[e2] & (NPERN - 1);
    atomicAdd(&Bm[r * LB + c], 1.0f);
  }
  __syncthreads();

  // ---- degrees, dinv = (rowsum + 1e-6)^-1/2 ---------------------------
  if (tid < NPERN) {
    float s = 1e-6f;
    for (int j = 0; j < NPERN; ++j) s += Bm[tid * LB + j];
    dinv[tid] = rsqrtf(s);
  }
  __syncthreads();

  // ---- B = 2I - L = I + Dinv A Dinv (in place) ------------------------
  for (int idx = tid; idx < NPERN * NPERN; idx += 128) {
    int i = idx >> 6, j = idx & 63;
    float v = dinv[i] * Bm[i * LB + j] * dinv[j];
    if (i == j) v += 1.0f;
    Bm[i * LB + j] = v;
  }

  // ---- deterministic pseudo-random init of X --------------------------
  for (int idx = tid; idx < NPERN * MSUB; idx += 128) {
    int i = idx >> 5, j = idx & 31;
    unsigned h = (unsigned)(i * 131 + j * 71 + 13) * 2654435761u;
    float r = (float)((h >> 9) & 0xFFFFu) * (1.0f / 65536.0f) - 0.5f;
    Xl[i * LX + j] = r + ((i == j) ? 1.0f : 0.0f);
  }
  __syncthreads();

  // ---- subspace iteration: X <- orth(B @ X) via CholeskyQR ------------
  for (int it = 0; it < PITER; ++it) {
    {  // Y = B @ X : wave wv owns rows [wv*16, wv*16+16), both N halves
      int m0 = wv * 16;
      v8f a0 =
          wmma_mm16<false, false>(&Bm[m0 * LB], LB, &Xl[0], LX, NPERN, lane);
      v8f a1 =
          wmma_mm16<false, false>(&Bm[m0 * LB], LB, &Xl[16], LX, NPERN, lane);
      store_tile(&Yl[m0 * LX], LX, lane, a0);
      store_tile(&Yl[m0 * LX + 16], LX, lane, a1);
    }
    __syncthreads();
    {  // Gram matrix G = Y^T Y -> Sl; wave wv -> tile (wv>>1, wv&1)
      int mi = (wv >> 1) * 16, ni = (wv & 1) * 16;
      v8f acc = wmma_mm16<true, false>(&Yl[mi], LX, &Yl[ni], LX, NPERN, lane);
      store_tile(&Sl[mi * LS + ni], LS, lane, acc);
    }
    __syncthreads();
    // Left-looking Cholesky G = L L^T.
    // Off-diagonal L[i][k] (i>k) overwrites Sl[i][k]; diagonal kept as
    // invd[k] = 1/L[k][k] so Sl diag entries are never raced.
    for (int k = 0; k < MSUB; ++k) {
      if (tid < MSUB && tid >= k) {
        float s = Sl[tid * LS + k];   // G[i][k], own element
        float dd = Sl[k * LS + k];    // G[k][k], never overwritten
        for (int m = 0; m < k; ++m) {
          float lkm = Sl[k * LS + m];  // finalized column m (row k)
          s -= Sl[tid * LS + m] * lkm;
          dd -= lkm * lkm;
        }
        float di = rsqrtf(fmaxf(dd, 1e-12f));
        if (tid == k)
          invd[k] = di;
        else
          Sl[tid * LS + k] = s * di;
      }
      __syncthreads();
    }
    // Minv = L^-1 (lower) by forward substitution; columns independent,
    // one thread per column, written into Wl (free at this point).
    if (tid < MSUB) {
      const int j = tid;
      Wl[j * LS + j] = invd[j];
      for (int i = j + 1; i < MSUB; ++i) {
        float acc = 0.0f;
        for (int m = j; m < i; ++m) acc += Sl[i * LS + m] * Wl[m * LS + j];
        Wl[i * LS + j] = -acc * invd[i];
      }
    }
    __syncthreads();
    {  // X = Y @ Minv^T  (Q factor of Y); transposed-B WMMA load
      int m0 = wv * 16;
      v8f a0 =
          wmma_mm16<false, true>(&Yl[m0 * LX], LX, &Wl[0], LS, MSUB, lane);
      v8f a1 = wmma_mm16<false, true>(&Yl[m0 * LX], LX, &Wl[16 * LS], LS,
                                      MSUB, lane);
      store_tile(&Xl[m0 * LX], LX, lane, a0);
      store_tile(&Xl[m0 * LX + 16], LX, lane, a1);
    }
    __syncthreads();
  }

  // ---- Rayleigh-Ritz: S = X^T (B X) -----------------------------------
  {
    int m0 = wv * 16;
    v8f a0 =
        wmma_mm16<false, false>(&Bm[m0 * LB], LB, &Xl[0], LX, NPERN, lane);
    v8f a1 =
        wmma_mm16<false, false>(&Bm[m0 * LB], LB, &Xl[16], LX, NPERN, lane);
    store_tile(&Yl[m0 * LX], LX, lane, a0);
    store_tile(&Yl[m0 * LX + 16], LX, lane, a1);
  }
  __syncthreads();
  {  // 2x2 tile grid; wave wv -> tile (wv>>1, wv&1); A = X^T
    int mi = (wv >> 1) * 16, ni = (wv & 1) * 16;
    v8f acc = wmma_mm16<true, false>(&Xl[mi], LX, &Yl[ni], LX, NPERN, lane);
    store_tile(&Sl[mi * LS + ni], LS, lane, acc);
  }
  __syncthreads();
  // symmetrize S, init W = I (overwrites Minv scratch)
  for (int idx = tid; idx < MSUB * MSUB; idx += 128) {
    int i = idx >> 5, j = idx & 31;
    Wl[i * LS + j] = (i == j) ? 1.0f : 0.0f;
    if (j > i) {
      float v = 0.5f * (Sl[i * LS + j] + Sl[j * LS + i]);
      Sl[i * LS + j] = v;
      Sl[j * LS + i] = v;
    }
  }
  __syncthreads();

  // ---- parallel-ordered Jacobi on 32x32 S, accumulate W ---------------
  for (int sw = 0; sw < JSWEEPS; ++sw) {
    for (int r = 0; r < 31; ++r) {
      if (tid < 16) {  // round-robin tournament pairing: 16 disjoint pairs
        int p = (tid == 0) ? 0 : 1 + ((tid - 1 + r) % 31);
        int q = 1 + ((30 - tid + r) % 31);
        if (p > q) { int t2 = p; p = q; q = t2; }
        pb[tid] = p;
        qb[tid] = q;
        float app = Sl[p * LS + p];
        float aqq = Sl[q * LS + q];
        float apq = Sl[p * LS + q];
        float c = 1.0f, s = 0.0f;
        if (fabsf(apq) > 1e-9f) {
          float tau = (aqq - app) / (2.0f * apq);
          float t3 = ((tau >= 0.0f) ? 1.0f : -1.0f) /
                     (fabsf(tau) + sqrtf(1.0f + tau * tau));
          c = rsqrtf(1.0f + t3 * t3);
          s = t3 * c;
        }
        cb[tid] = c;
        sb[tid] = s;
      }
      __syncthreads();
      {  // column update of S and W: 8 threads per pair, 4 rows each
        int k = tid >> 3, rb = tid & 7;
        int p = pb[k], q = qb[k];
        float c = cb[k], s = sb[k];
#pragma unroll
        for (int m = 0; m < 4; ++m) {
          int i = rb + 8 * m;
          float sip = Sl[i * LS + p], siq = Sl[i * LS + q];
          Sl[i * LS + p] = c * sip - s * siq;
          Sl[i * LS + q] = s * sip + c * siq;
          float wip = Wl[i * LS + p], wiq = Wl[i * LS + q];
          Wl[i * LS + p] = c * wip - s * wiq;
          Wl[i * LS + q] = s * wip + c * wiq;
        }
      }
      __syncthreads();
      {  // row update of S
        int k = tid >> 3, cbse = tid & 7;
        int p = pb[k], q = qb[k];
        float c = cb[k], s = sb[k];
#pragma unroll
        for (int m = 0; m < 4; ++m) {
          int i = cbse + 8 * m;
          float spi = Sl[p * LS + i], sqi = Sl[q * LS + i];
          Sl[p * LS + i] = c * spi - s * sqi;
          Sl[q * LS + i] = s * spi + c * sqi;
        }
      }
      __syncthreads();
    }
  }

  // ---- pick 16 largest Ritz values of B (= smallest eig of L, ascending)
  if (tid == 0) {
    unsigned used = 0u;
    for (int j = 0; j < KOUT; ++j) {
      int best = 0;
      float bv = -1e30f;
      for (int i = 0; i < MSUB; ++i) {
        if (((used >> i) & 1u) == 0u) {
          float v = Sl[i * LS + i];
          if (v > bv) { bv = v; best = i; }
        }
      }
      used |= (1u << best);
      ord[j] = best;
    }
  }
  __syncthreads();
  for (int idx = tid; idx < MSUB * KOUT; idx += 128) {
    int i = idx >> 4, j = idx & 15;
    Wsel[i * KOUT + j] = Wl[i * LS + ord[j]];
  }
  __syncthreads();

  // ---- P = X @ Wsel  (64x32 @ 32x16) -> global ------------------------
  {
    int m0 = wv * 16;
    v8f acc =
        wmma_mm16<false, false>(&Xl[m0 * LX], LX, &Wsel[0], KOUT, MSUB, lane);
    const int mr = lane & 15;
    const int hi = (lane >> 4) & 1;
    float* op = out + (size_t)g * NPERN * KOUT;
#pragma unroll
    for (int rr = 0; rr < 8; ++rr)
      op[(m0 + rr + 8 * hi) * KOUT + mr] = acc[rr];
  }
}

extern "C" void kernel_launch(void* const* d_in, const int* in_sizes, int n_in,
                              void* d_out, int out_size, void* d_ws,
                              size_t ws_size, hipStream_t stream) {
  (void)in_sizes; (void)n_in; (void)d_ws; (void)ws_size; (void)out_size;
  const int* edge_index = (const int*)d_in[0];  // [2, 2*G*EPER] int32
  float* out = (float*)d_out;                   // [G*NPER, K] float32
  lappe_eigh_kernel<<<GG, 128, 0, stream>>>(edge_index, out);
}